// GAT_5970004541990
// MI455X (gfx1250) — compile-verified
//
#include <hip/hip_runtime.h>
#include <math.h>

// ---------------------------------------------------------------------------
// CDNA5 (gfx1250) GAT: 2-layer GATConv + mean pool + log_softmax.
// Dense feature transforms use V_WMMA_F32_16X16X4_F32 (fp32 WMMA, wave32).
// ---------------------------------------------------------------------------

typedef float v2f __attribute__((ext_vector_type(2)));
typedef float v8f __attribute__((ext_vector_type(8)));

#define NEG_SLOPE 0.2f

// ---------------------------------------------------------------------------
// GEMM C[M,N] = A[M,K] * B[K,N], all row-major fp32.
// One wave computes one 16x16 output tile via V_WMMA_F32_16X16X4_F32.
// Requires: M%16==0, N%16==0, K%4==0. 256 threads = 8 waves per block.
// VGPR layouts (ISA 7.12.2):
//   A 16x4 : lane(0-15)=M, vgpr0/1 = K{0,1}; lanes 16-31 same M, K{2,3}
//   B 4x16 : lane%16 = N, vgpr0/1 = K{0,1} (lanes 0-15) / K{2,3} (lanes 16-31)
//   C 16x16: vgpr r -> M = r + 8*(lane>=16), N = lane%16
// ---------------------------------------------------------------------------
__global__ void __launch_bounds__(256) gemm_f32_wmma(
    const float* __restrict__ A, const float* __restrict__ B,
    float* __restrict__ C, int M, int K, int N)
{
    const int lane = threadIdx.x & 31;
    const int wave = threadIdx.x >> 5;
    const int half = lane >> 4;        // 0 or 1
    const int l16  = lane & 15;

    const int tilesN = N >> 4;
    const int tilesM = M >> 4;
    const int tile   = blockIdx.x * 8 + wave;   // uniform within wave
    if (tile >= tilesM * tilesN) return;

    const int tm = tile / tilesN;
    const int tn = tile - tm * tilesN;
    const int m0 = tm << 4;
    const int n0 = tn << 4;

    const float* __restrict__ Arow = A + (size_t)(m0 + l16) * K;
    const float* __restrict__ Bcol = B + (size_t)(n0 + l16);

    v8f acc = {};
#pragma unroll 4
    for (int k0 = 0; k0 < K; k0 += 4) {
        const int ka = k0 + 2 * half;
        v2f a, b;
        a.x = Arow[ka + 0];
        a.y = Arow[ka + 1];
        b.x = Bcol[(size_t)(ka + 0) * N];
        b.y = Bcol[(size_t)(ka + 1) * N];
        // 8 args: (neg_a, A, neg_b, B, c_mod, C, reuse_a, reuse_b)
        acc = __builtin_amdgcn_wmma_f32_16x16x4_f32(
            false, a, false, b, (short)0, acc, false, false);
    }

#pragma unroll
    for (int r = 0; r < 8; ++r) {
        C[(size_t)(m0 + r + 8 * half) * N + (n0 + l16)] = acc[r];
    }
}

// ---------------------------------------------------------------------------
// Per-node attention logits: als[n] = h[n,:].a_src, ald[n] = h[n,:].a_dst
// One wave32 per node, shuffle reduction.
// ---------------------------------------------------------------------------
__global__ void __launch_bounds__(256) node_logits(
    const float* __restrict__ h, const float* __restrict__ a_s,
    const float* __restrict__ a_d, float* __restrict__ als,
    float* __restrict__ ald, int N, int D)
{
    const int lane = threadIdx.x & 31;
    const int node = (blockIdx.x * blockDim.x + threadIdx.x) >> 5;
    if (node >= N) return;
    const float* __restrict__ hp = h + (size_t)node * D;
    float vs = 0.f, vd = 0.f;
    for (int f = lane; f < D; f += 32) {
        const float x = hp[f];
        vs += x * a_s[f];
        vd += x * a_d[f];
    }
#pragma unroll
    for (int off = 16; off; off >>= 1) {
        vs += __shfl_xor(vs, off, 32);
        vd += __shfl_xor(vd, off, 32);
    }
    if (lane == 0) { als[node] = vs; ald[node] = vd; }
}

// ---------------------------------------------------------------------------
// Helpers: edge fetch (self-loops appended), order-preserving f32<->i32.
// ---------------------------------------------------------------------------
__device__ __forceinline__ void get_edge(const int* __restrict__ ei, int E,
                                         int i, int& s, int& d)
{
    if (i < E) { s = ei[i]; d = ei[(size_t)E + i]; }
    else       { s = d = i - E; }
}

__device__ __forceinline__ int   enc_f(float f) {
    int i = __float_as_int(f);
    return (i < 0) ? (i ^ 0x7FFFFFFF) : i;
}
__device__ __forceinline__ float dec_f(int i) {
    return __int_as_float((i < 0) ? (i ^ 0x7FFFFFFF) : i);
}
__device__ __forceinline__ float lrelu(float e) {
    return e > 0.f ? e : NEG_SLOPE * e;
}

// Pass 1: segment max over destination nodes (atomic on encoded ints).
__global__ void __launch_bounds__(256) edge_max(
    const int* __restrict__ ei, const float* __restrict__ als,
    const float* __restrict__ ald, int* __restrict__ menc, int E, int Etot)
{
    const int i = blockIdx.x * blockDim.x + threadIdx.x;
    if (i >= Etot) return;
    int s, d; get_edge(ei, E, i, s, d);
    atomicMax(&menc[d], enc_f(lrelu(als[s] + ald[d])));
}

// Pass 2: segment sum of exp(e - m[dst]).
__global__ void __launch_bounds__(256) edge_sum(
    const int* __restrict__ ei, const float* __restrict__ als,
    const float* __restrict__ ald, const int* __restrict__ menc,
    float* __restrict__ ssum, int E, int Etot)
{
    const int i = blockIdx.x * blockDim.x + threadIdx.x;
    if (i >= Etot) return;
    int s, d; get_edge(ei, E, i, s, d);
    const float e = lrelu(als[s] + ald[d]);
    atomicAdd(&ssum[d], expf(e - dec_f(menc[d])));
}

// Pass 3: out[dst,:] += alpha * h[src,:]. One wave per edge.
__global__ void __launch_bounds__(256) edge_aggregate(
    const int* __restrict__ ei, const float* __restrict__ als,
    const float* __restrict__ ald, const int* __restrict__ menc,
    const float* __restrict__ ssum, const float* __restrict__ h,
    float* __restrict__ out, int E, int Etot, int D)
{
    const int lane = threadIdx.x & 31;
    const int i = (blockIdx.x * blockDim.x + threadIdx.x) >> 5;
    if (i >= Etot) return;
    int s, d; get_edge(ei, E, i, s, d);
    const float e = lrelu(als[s] + ald[d]);
    const float alpha = expf(e - dec_f(menc[d])) / ssum[d];
    const float* __restrict__ hp = h + (size_t)s * D;
    float* __restrict__ op = out + (size_t)d * D;
    for (int f = lane; f < D; f += 32) {
        atomicAdd(&op[f], alpha * hp[f]);
    }
}

// out[n,f] = (relu?)(out[n,f] + b[f])
__global__ void __launch_bounds__(256) bias_act(
    float* __restrict__ out, const float* __restrict__ b, int N, int D, int relu)
{
    const int i = blockIdx.x * blockDim.x + threadIdx.x;
    if (i >= N * D) return;
    const int f = i % D;
    float v = out[i] + b[f];
    if (relu) v = v > 0.f ? v : 0.f;
    out[i] = v;
}

__global__ void __launch_bounds__(256) fill_f32(float* __restrict__ p, float v, int n)
{
    const int i = blockIdx.x * blockDim.x + threadIdx.x;
    if (i < n) p[i] = v;
}
__global__ void __launch_bounds__(256) fill_i32(int* __restrict__ p, int v, int n)
{
    const int i = blockIdx.x * blockDim.x + threadIdx.x;
    if (i < n) p[i] = v;
}

// Mean-pool accumulation: pooled[batch[n], f] += h[n, f]; cnt[g] += 1.
__global__ void __launch_bounds__(256) pool_accum(
    const float* __restrict__ h, const int* __restrict__ batch,
    float* __restrict__ pooled, float* __restrict__ cnt, int N, int D)
{
    const int i = blockIdx.x * blockDim.x + threadIdx.x;
    if (i >= N * D) return;
    const int node = i / D;
    const int f = i - node * D;
    const int g = batch[node];
    atomicAdd(&pooled[g * D + f], h[i]);
    if (f == 0) atomicAdd(&cnt[g], 1.0f);
}

// Final: out[g,:] = log_softmax(pooled[g,:]/max(cnt,1)). One thread per graph.
__global__ void __launch_bounds__(64) pool_logsoftmax(
    const float* __restrict__ pooled, const float* __restrict__ cnt,
    float* __restrict__ out, int G)
{
    const int g = blockIdx.x * blockDim.x + threadIdx.x;
    if (g >= G) return;
    float c = cnt[g];
    c = c > 1.f ? c : 1.f;
    float vals[16];
    float m = -INFINITY;
#pragma unroll
    for (int k = 0; k < 16; ++k) {
        vals[k] = pooled[g * 16 + k] / c;
        m = fmaxf(m, vals[k]);
    }
    float s = 0.f;
#pragma unroll
    for (int k = 0; k < 16; ++k) s += expf(vals[k] - m);
    const float ls = logf(s);
#pragma unroll
    for (int k = 0; k < 16; ++k) out[g * 16 + k] = vals[k] - m - ls;
}

// ---------------------------------------------------------------------------
// Host-side orchestration (graph-capture safe: only kernel launches on stream)
// ---------------------------------------------------------------------------
extern "C" void kernel_launch(void* const* d_in, const int* in_sizes, int n_in,
                              void* d_out, int out_size, void* d_ws, size_t ws_size,
                              hipStream_t stream)
{
    const float* x    = (const float*)d_in[0];
    const int*   ei   = (const int*)  d_in[1];   // [2, E] row-major
    const int*   batch= (const int*)  d_in[2];
    const float* W1   = (const float*)d_in[3];
    const float* a1s  = (const float*)d_in[4];
    const float* a1d  = (const float*)d_in[5];
    const float* b1   = (const float*)d_in[6];
    const float* W2   = (const float*)d_in[7];
    const float* a2s  = (const float*)d_in[8];
    const float* a2d  = (const float*)d_in[9];
    const float* b2   = (const float*)d_in[10];
    float* out = (float*)d_out;

    const int N = 50000, E = 640000, G = 64;
    const int Din = 128, Dh = 128, Do = 16;
    const int Etot = E + N;  // self loops appended

    // Workspace layout (floats)
    float* ws   = (float*)d_ws;
    float* h1   = ws;  ws += (size_t)N * Dh;   // x @ W1
    float* o1   = ws;  ws += (size_t)N * Dh;   // layer-1 output (relu'd in place)
    float* h2   = ws;  ws += (size_t)N * Do;   // o1 @ W2
    float* o2   = ws;  ws += (size_t)N * Do;   // layer-2 output
    float* als  = ws;  ws += N;
    float* ald  = ws;  ws += N;
    int*   menc = (int*)ws; ws += N;
    float* ssum = ws;  ws += N;
    float* pooled = ws; ws += G * Do;
    float* cnt  = ws;  ws += G;

    const int TB = 256;
    const int gNodeWarp = (N + 7) / 8;          // warp per node
    const int gEdge     = (Etot + TB - 1) / TB; // thread per edge
    const int gEdgeWarp = (Etot + 7) / 8;       // warp per edge
    const int gNDh      = (N * Dh + TB - 1) / TB;
    const int gNDo      = (N * Do + TB - 1) / TB;
    const int gN        = (N + TB - 1) / TB;

    // ---------------- Layer 1 ----------------
    {
        const int tiles = (N / 16) * (Dh / 16);
        gemm_f32_wmma<<<(tiles + 7) / 8, TB, 0, stream>>>(x, W1, h1, N, Din, Dh);
    }
    node_logits<<<gNodeWarp, TB, 0, stream>>>(h1, a1s, a1d, als, ald, N, Dh);
    fill_i32<<<gN, TB, 0, stream>>>(menc, (int)0x80000000, N);
    fill_f32<<<gN, TB, 0, stream>>>(ssum, 0.f, N);
    fill_f32<<<gNDh, TB, 0, stream>>>(o1, 0.f, N * Dh);
    edge_max<<<gEdge, TB, 0, stream>>>(ei, als, ald, menc, E, Etot);
    edge_sum<<<gEdge, TB, 0, stream>>>(ei, als, ald, menc, ssum, E, Etot);
    edge_aggregate<<<gEdgeWarp, TB, 0, stream>>>(ei, als, ald, menc, ssum, h1, o1, E, Etot, Dh);
    bias_act<<<gNDh, TB, 0, stream>>>(o1, b1, N, Dh, /*relu=*/1);

    // ---------------- Layer 2 ----------------
    {
        const int tiles = (N / 16) * (Do / 16);
        gemm_f32_wmma<<<(tiles + 7) / 8, TB, 0, stream>>>(o1, W2, h2, N, Dh, Do);
    }
    node_logits<<<gNodeWarp, TB, 0, stream>>>(h2, a2s, a2d, als, ald, N, Do);
    fill_i32<<<gN, TB, 0, stream>>>(menc, (int)0x80000000, N);
    fill_f32<<<gN, TB, 0, stream>>>(ssum, 0.f, N);
    fill_f32<<<gNDo, TB, 0, stream>>>(o2, 0.f, N * Do);
    edge_max<<<gEdge, TB, 0, stream>>>(ei, als, ald, menc, E, Etot);
    edge_sum<<<gEdge, TB, 0, stream>>>(ei, als, ald, menc, ssum, E, Etot);
    edge_aggregate<<<gEdgeWarp, TB, 0, stream>>>(ei, als, ald, menc, ssum, h2, o2, E, Etot, Do);
    bias_act<<<gNDo, TB, 0, stream>>>(o2, b2, N, Do, /*relu=*/0);

    // ---------------- Pool + log_softmax ----------------
    fill_f32<<<1, TB, 0, stream>>>(pooled, 0.f, G * Do);
    fill_f32<<<1, TB, 0, stream>>>(cnt, 0.f, G);
    pool_accum<<<gNDo, TB, 0, stream>>>(o2, batch, pooled, cnt, N, Do);
    pool_logsoftmax<<<1, 64, 0, stream>>>(pooled, cnt, out, G);
}